// DelayGNNStage_9268539425223
// MI455X (gfx1250) — compile-verified
//
#include <hip/hip_runtime.h>
#include <hip/hip_bf16.h>

typedef __attribute__((ext_vector_type(2))) float v2f;
typedef __attribute__((ext_vector_type(8))) float v8f;

constexpr int DD = 128;   // feature dim (fixed by problem)

// ---------------------------------------------------------------------------
// Degree accumulation: one thread per edge, hardware float atomic increments.
// ---------------------------------------------------------------------------
__global__ __launch_bounds__(256) void degree_k(const int* __restrict__ src,
                                                const int* __restrict__ dst,
                                                float* __restrict__ degOut,
                                                float* __restrict__ degIn,
                                                int nedges) {
    int i = blockIdx.x * blockDim.x + threadIdx.x;
    if (i >= nedges) return;
    unsafeAtomicAdd(&degOut[src[i]], 1.0f);
    unsafeAtomicAdd(&degIn[dst[i]], 1.0f);
}

// deg -> (deg > 0 ? rsqrt(deg) : 0), in place over 4*N entries
__global__ __launch_bounds__(256) void invdeg_k(float* __restrict__ deg, int n) {
    int i = blockIdx.x * blockDim.x + threadIdx.x;
    if (i >= n) return;
    float v = deg[i];
    deg[i] = (v > 0.0f) ? rsqrtf(v) : 0.0f;
}

__global__ __launch_bounds__(256) void edgenorm_k(const int* __restrict__ src,
                                                  const int* __restrict__ dst,
                                                  const float* __restrict__ invOut,
                                                  const float* __restrict__ invIn,
                                                  float* __restrict__ norm,
                                                  int nedges) {
    int i = blockIdx.x * blockDim.x + threadIdx.x;
    if (i >= nedges) return;
    norm[i] = invOut[src[i]] * invIn[dst[i]];
}

// ---------------------------------------------------------------------------
// H = X @ W  (X: nrows x 128 row-major, W: 128 x 128 row-major)
// Block = 8 waves, owns a 64-row x 128-col output tile.
//   - X slab (64x128 f32 = 32 KB) staged once into LDS, stride-padded to 132
//     floats so A-fragment ds_load_b64s are bank-conflict-free (132 % 64 = 4).
//   - Wave j owns columns [16j,16j+16); keeps 4 accumulators (rows i*16..).
//   - Each B fragment (2 scalar loads from one 64B cacheline of L2-resident W)
//     is reused by 4 independent v_wmma_f32_16x16x4_f32 ops.
//
// Fragment layouts per CDNA5 ISA (32-bit, K=4):
//   A: lane l, m=l&15: lanes 0-15 -> K={k0,k0+1}; lanes 16-31 -> K={k0+2,k0+3}
//   B: lane l, n=col:  lanes 0-15 -> rows {k0,k0+1}; lanes 16-31 -> {k0+2,k0+3}
//   C/D: vgpr r holds row (r + 8*(lane>=16)), col (lane&15).
// ---------------------------------------------------------------------------
__global__ __launch_bounds__(256) void gemm_xw(const float* __restrict__ X,
                                               const float* __restrict__ W,
                                               float* __restrict__ H,
                                               int nrows) {
    constexpr int LDA = DD + 4;                 // 132-float padded row stride
    __shared__ float ldsA[64 * LDA];

    const int wave = threadIdx.x >> 5;          // 0..7 -> column tile
    const int lane = threadIdx.x & 31;
    const int m    = lane & 15;
    const int hi   = lane >> 4;                 // 0 or 1
    const int n    = (wave << 4) + m;           // output column
    const int rowBase = blockIdx.x * 64;

    // ---- stage 64x128 X slab into LDS (coalesced float4, 8 iters/thread) ----
    for (int idx = threadIdx.x; idx < 64 * (DD / 4); idx += 256) {
        int r  = idx >> 5;                      // 32 float4 per row
        int c4 = idx & 31;
        int gr = rowBase + r;
        if (gr >= nrows) gr = nrows - 1;        // clamp; stores guarded later
        float4 v = *(const float4*)(X + (size_t)gr * DD + c4 * 4);
        *(float4*)(&ldsA[r * LDA + c4 * 4]) = v;  // 16B aligned (LDA%4==0)
    }
    __syncthreads();

    v8f c0 = {}, c1 = {}, c2 = {}, c3 = {};
#pragma unroll
    for (int k0 = 0; k0 < DD; k0 += 4) {
        const int ka = k0 + (hi << 1);
        v2f b;
        b.x = W[ka * DD + n];
        b.y = W[(ka + 1) * DD + n];
        v2f a0 = *(const v2f*)(&ldsA[(0 * 16 + m) * LDA + ka]);
        v2f a1 = *(const v2f*)(&ldsA[(1 * 16 + m) * LDA + ka]);
        v2f a2 = *(const v2f*)(&ldsA[(2 * 16 + m) * LDA + ka]);
        v2f a3 = *(const v2f*)(&ldsA[(3 * 16 + m) * LDA + ka]);
        c0 = __builtin_amdgcn_wmma_f32_16x16x4_f32(false, a0, false, b, (short)0, c0, false, false);
        c1 = __builtin_amdgcn_wmma_f32_16x16x4_f32(false, a1, false, b, (short)0, c1, false, false);
        c2 = __builtin_amdgcn_wmma_f32_16x16x4_f32(false, a2, false, b, (short)0, c2, false, false);
        c3 = __builtin_amdgcn_wmma_f32_16x16x4_f32(false, a3, false, b, (short)0, c3, false, false);
    }

    // ---- store 4 sub-tiles ----
    v8f acc[4] = { c0, c1, c2, c3 };
#pragma unroll
    for (int i = 0; i < 4; ++i) {
        int rb = rowBase + i * 16 + (hi << 3);
#pragma unroll
        for (int r = 0; r < 8; ++r) {
            int orow = rb + r;
            if (orow < nrows) H[(size_t)orow * DD + n] = acc[i][r];
        }
    }
}

// ---------------------------------------------------------------------------
// Scatter: one wave32 per edge; lane handles 4 consecutive features (float4).
// S[dst] += H[src] * norm  via hardware global_atomic_add_f32.
// ---------------------------------------------------------------------------
__global__ __launch_bounds__(256) void scatter_edges(const float* __restrict__ H,
                                                     const int* __restrict__ src,
                                                     const int* __restrict__ dst,
                                                     const float* __restrict__ norm,
                                                     float* __restrict__ S,
                                                     int nedges) {
    int w    = (blockIdx.x * blockDim.x + threadIdx.x) >> 5;
    int lane = threadIdx.x & 31;
    if (w >= nedges) return;
    int s = src[w], d = dst[w];
    float nw = norm[w];
    float4 v = ((const float4*)(H + (size_t)s * DD))[lane];
    float* o = S + (size_t)d * DD + lane * 4;
    unsafeAtomicAdd(o + 0, v.x * nw);
    unsafeAtomicAdd(o + 1, v.y * nw);
    unsafeAtomicAdd(o + 2, v.z * nw);
    unsafeAtomicAdd(o + 3, v.w * nw);
}

// ---------------------------------------------------------------------------
// Update: one wave32 per node.
//   a = softmax(alpha[t])  (a0=1,a1=0 for layer 0)
//   agg = a0*(S1+b1) + a1*(S2+b2);  xn = x + relu(agg);  xn /= max(||xn||2,1e-12)
// ---------------------------------------------------------------------------
__global__ __launch_bounds__(256) void update_k(const float* __restrict__ xcur,
                                                const float* __restrict__ S1,
                                                const float* __restrict__ S2,
                                                const float* __restrict__ b1,
                                                const float* __restrict__ b2,
                                                const float* __restrict__ alpha,
                                                int hop2,
                                                float* __restrict__ xnext,
                                                int nnodes) {
    int w    = (blockIdx.x * blockDim.x + threadIdx.x) >> 5;
    int lane = threadIdx.x & 31;
    if (w >= nnodes) return;

    float a0 = 1.0f, a1 = 0.0f;
    if (hop2) {
        float r0 = alpha[0], r1 = alpha[1];
        float mx = fmaxf(r0, r1);
        float e0 = expf(r0 - mx), e1 = expf(r1 - mx);
        float inv = 1.0f / (e0 + e1);
        a0 = e0 * inv;
        a1 = e1 * inv;
    }

    size_t base = (size_t)w * DD + lane * 4;
    float4 x  = *(const float4*)(xcur + base);
    float4 s1 = *(const float4*)(S1 + base);
    float4 c1 = *(const float4*)(b1 + lane * 4);

    float4 agg;
    agg.x = a0 * (s1.x + c1.x);
    agg.y = a0 * (s1.y + c1.y);
    agg.z = a0 * (s1.z + c1.z);
    agg.w = a0 * (s1.w + c1.w);
    if (hop2) {
        float4 s2 = *(const float4*)(S2 + base);
        float4 c2 = *(const float4*)(b2 + lane * 4);
        agg.x += a1 * (s2.x + c2.x);
        agg.y += a1 * (s2.y + c2.y);
        agg.z += a1 * (s2.z + c2.z);
        agg.w += a1 * (s2.w + c2.w);
    }

    float4 xn;
    xn.x = x.x + fmaxf(agg.x, 0.0f);
    xn.y = x.y + fmaxf(agg.y, 0.0f);
    xn.z = x.z + fmaxf(agg.z, 0.0f);
    xn.w = x.w + fmaxf(agg.w, 0.0f);

    float ss = xn.x * xn.x + xn.y * xn.y + xn.z * xn.z + xn.w * xn.w;
#pragma unroll
    for (int off = 16; off >= 1; off >>= 1)
        ss += __shfl_xor(ss, off, 32);

    float inv = 1.0f / fmaxf(sqrtf(ss), 1e-12f);
    xn.x *= inv; xn.y *= inv; xn.z *= inv; xn.w *= inv;
    *(float4*)(xnext + base) = xn;
}

// ---------------------------------------------------------------------------
extern "C" void kernel_launch(void* const* d_in, const int* in_sizes, int n_in,
                              void* d_out, int out_size, void* d_ws, size_t ws_size,
                              hipStream_t stream) {
    const float* x     = (const float*)d_in[0];
    const float* W1    = (const float*)d_in[1];   // [L,D,D]
    const float* b1    = (const float*)d_in[2];   // [L,D]
    const float* W2    = (const float*)d_in[3];   // [L,D,D]
    const float* b2    = (const float*)d_in[4];   // [L,D]
    const float* alpha = (const float*)d_in[5];   // [L,2]
    const int*   src1  = (const int*)d_in[6];
    const int*   dst1  = (const int*)d_in[7];
    const int*   src2  = (const int*)d_in[8];
    const int*   dst2  = (const int*)d_in[9];

    const int Lnum = in_sizes[5] / 2;             // 4
    const int N    = in_sizes[0] / DD;            // 50000
    const int E1   = in_sizes[6];
    const int E2   = in_sizes[8];
    const size_t ND = (size_t)N * DD;

    // workspace layout (floats)
    float* ws      = (float*)d_ws;
    float* degOut1 = ws;
    float* degIn1  = ws + (size_t)N;
    float* degOut2 = ws + 2 * (size_t)N;
    float* degIn2  = ws + 3 * (size_t)N;
    float* norm1   = ws + 4 * (size_t)N;
    float* norm2   = norm1 + E1;
    float* H1      = norm2 + E2;
    float* H2      = H1 + ND;
    float* S1      = H2 + ND;
    float* S2      = S1 + ND;
    float* bufB    = S2 + ND;
    float* bufC    = bufB + ND;

    // ---- edge norms ----
    hipMemsetAsync(ws, 0, 4 * (size_t)N * sizeof(float), stream);
    degree_k<<<(E1 + 255) / 256, 256, 0, stream>>>(src1, dst1, degOut1, degIn1, E1);
    degree_k<<<(E2 + 255) / 256, 256, 0, stream>>>(src2, dst2, degOut2, degIn2, E2);
    invdeg_k<<<(4 * N + 255) / 256, 256, 0, stream>>>(ws, 4 * N);
    edgenorm_k<<<(E1 + 255) / 256, 256, 0, stream>>>(src1, dst1, degOut1, degIn1, norm1, E1);
    edgenorm_k<<<(E2 + 255) / 256, 256, 0, stream>>>(src2, dst2, degOut2, degIn2, norm2, E2);

    // ---- layer loop: xs[0]=x, nexts: t0->bufB, t1->bufC, t2->bufB, t3->d_out
    // (t2 may reuse bufB: its GEMM read of prev=bufB precedes the update write
    //  in stream order, and update only reads cur=bufC.)
    float* nexts[4] = { bufB, bufC, bufB, (float*)d_out };
    const float* cur  = x;
    const float* prev = nullptr;

    const int gemmBlocks     = (N + 63) / 64;
    const int scatterBlocks1 = (int)(((size_t)E1 * 32 + 255) / 256);
    const int scatterBlocks2 = (int)(((size_t)E2 * 32 + 255) / 256);
    const int updateBlocks   = (int)(((size_t)N * 32 + 255) / 256);

    for (int t = 0; t < Lnum; ++t) {
        gemm_xw<<<gemmBlocks, 256, 0, stream>>>(cur, W1 + (size_t)t * DD * DD, H1, N);
        if (t > 0)
            gemm_xw<<<gemmBlocks, 256, 0, stream>>>(prev, W2 + (size_t)t * DD * DD, H2, N);

        hipMemsetAsync(S1, 0, ND * sizeof(float), stream);
        scatter_edges<<<scatterBlocks1, 256, 0, stream>>>(H1, src1, dst1, norm1, S1, E1);
        if (t > 0) {
            hipMemsetAsync(S2, 0, ND * sizeof(float), stream);
            scatter_edges<<<scatterBlocks2, 256, 0, stream>>>(H2, src2, dst2, norm2, S2, E2);
        }

        update_k<<<updateBlocks, 256, 0, stream>>>(cur, S1, S2,
                                                   b1 + (size_t)t * DD,
                                                   b2 + (size_t)t * DD,
                                                   alpha + (size_t)t * 2,
                                                   t > 0 ? 1 : 0,
                                                   nexts[t], N);
        prev = cur;
        cur  = nexts[t];
    }
}